// NavierStokesKernel_10393820857045
// MI455X (gfx1250) — compile-verified
//
#include <hip/hip_runtime.h>
#include <math.h>

typedef __attribute__((ext_vector_type(2))) float v2f;
typedef __attribute__((ext_vector_type(8))) float v8f;
typedef __attribute__((ext_vector_type(4))) int   v4i;

#define NN   1024
#define NMSK 1023
#define NSQ  (1 << 20)          // 1024*1024 elements per image
#define BB   8
#define TOT  (BB * NSQ)         // total field elements
#define DT_C 0.01f
#define NU_C 0.01f

// --- CDNA5 async global->LDS path (guarded so the file always compiles) ----
#if defined(__has_builtin)
#if __has_builtin(__builtin_amdgcn_global_load_async_to_lds_b128) && \
    __has_builtin(__builtin_amdgcn_s_wait_asynccnt)
#define NS_ASYNC 1
#endif
#endif
#ifndef NS_ASYNC
#define NS_ASYNC 0
#endif

// ---------------------------------------------------------------------------
// Stencil / pointwise kernels (bandwidth-bound, periodic BC via & 1023)
// ---------------------------------------------------------------------------

__global__ __launch_bounds__(256) void ns_gen_dft(float* __restrict__ C, float* __restrict__ S) {
    int idx = blockIdx.x * 256 + threadIdx.x;      // exactly 1<<20 threads
    int m = idx >> 10, n = idx & NMSK;
    // exp(-2*pi*i*m*n/N) is periodic in (m*n) mod N -> keep sin/cos args small
    float th = (float)((m * n) & NMSK) * (6.283185307179586f / 1024.0f);
    C[idx] = cosf(th);
    S[idx] = sinf(th);
}

__global__ __launch_bounds__(256) void ns_smooth(const float* __restrict__ fu_in, float* __restrict__ fu_out,
                                                 const float* __restrict__ fv_in, float* __restrict__ fv_out) {
    int idx = blockIdx.x * 256 + threadIdx.x;      // exactly TOT threads
    int b = idx >> 20, rem = idx & (NSQ - 1);
    int y = rem >> 10, x = rem & NMSK;
    int xm = (x + NMSK) & NMSK, xp = (x + 1) & NMSK;
    int ym = (y + NMSK) & NMSK, yp = (y + 1) & NMSK;
    size_t base = (size_t)b << 20;
    int c = (y << 10) + x;
    {
        float fc = fu_in[base + c];
        float lap = fu_in[base + (y << 10) + xm] + fu_in[base + (y << 10) + xp]
                  + fu_in[base + (ym << 10) + x] + fu_in[base + (yp << 10) + x] - 4.0f * fc;
        fu_out[base + c] = fc + 0.1f * lap;
    }
    {
        float fc = fv_in[base + c];
        float lap = fv_in[base + (y << 10) + xm] + fv_in[base + (y << 10) + xp]
                  + fv_in[base + (ym << 10) + x] + fv_in[base + (yp << 10) + x] - 4.0f * fc;
        fv_out[base + c] = fc + 0.1f * lap;
    }
}

__global__ __launch_bounds__(256) void ns_sumsq(const float* __restrict__ fu, const float* __restrict__ fv,
                                                float* __restrict__ norms) {
    __shared__ float sm[256];
    int b = blockIdx.y;
    size_t base = (size_t)b << 20;
    float su = 0.f, sv = 0.f;
    for (int i = blockIdx.x * 256 + threadIdx.x; i < NSQ; i += 256 * 256) {
        float a = fu[base + i]; su += a * a;
        float c = fv[base + i]; sv += c * c;
    }
    sm[threadIdx.x] = su; __syncthreads();
    for (int s = 128; s > 0; s >>= 1) { if (threadIdx.x < s) sm[threadIdx.x] += sm[threadIdx.x + s]; __syncthreads(); }
    if (threadIdx.x == 0) atomicAdd(&norms[b], sm[0]);
    __syncthreads();
    sm[threadIdx.x] = sv; __syncthreads();
    for (int s = 128; s > 0; s >>= 1) { if (threadIdx.x < s) sm[threadIdx.x] += sm[threadIdx.x + s]; __syncthreads(); }
    if (threadIdx.x == 0) atomicAdd(&norms[8 + b], sm[0]);
}

__global__ __launch_bounds__(256) void ns_ustar(const float* __restrict__ x, const float* __restrict__ beta,
                                                const float* __restrict__ fu, const float* __restrict__ fv,
                                                const float* __restrict__ norms,
                                                float* __restrict__ us, float* __restrict__ vs) {
    int idx = blockIdx.x * 256 + threadIdx.x;
    int b = idx >> 20, rem = idx & (NSQ - 1);
    int y = rem >> 10, xx = rem & NMSK;
    int xm = (xx + NMSK) & NMSK, xp = (xx + 1) & NMSK;
    int ym = (y + NMSK) & NMSK, yp = (y + 1) & NMSK;
    const float* u = x + (((size_t)b * 3) << 20);
    const float* v = u + NSQ;
    const float* p = v + NSQ;
    int c = (y << 10) + xx, cxp = (y << 10) + xp, cxm = (y << 10) + xm;
    int cyp = (yp << 10) + xx, cym = (ym << 10) + xx;
    float uc = u[c], uxp = u[cxp], uxm = u[cxm], uyp = u[cyp], uym = u[cym];
    float vc = v[c], vxp = v[cxp], vxm = v[cxm], vyp = v[cyp], vym = v[cym];
    float adv_u = uc * 0.5f * (uxp - uxm) + vc * 0.5f * (uyp - uym);
    float adv_v = uc * 0.5f * (vxp - vxm) + vc * 0.5f * (vyp - vym);
    float lap_u = uxp + uxm + uyp + uym - 4.f * uc;
    float lap_v = vxp + vxm + vyp + vym - 4.f * vc;
    float dpdx = 0.5f * (p[cxp] - p[cxm]);
    float dpdy = 0.5f * (p[cyp] - p[cym]);
    float sb  = sqrtf(beta[b]);
    float inu = 1.f / fmaxf(sqrtf(norms[b]),     1e-12f);
    float inv = 1.f / fmaxf(sqrtf(norms[8 + b]), 1e-12f);
    size_t base = (size_t)b << 20;
    us[base + c] = uc + DT_C * (-adv_u + NU_C * lap_u - dpdx) + sb * fu[base + c] * inu;
    vs[base + c] = vc + DT_C * (-adv_v + NU_C * lap_v - dpdy) + sb * fv[base + c] * inv;
}

__global__ __launch_bounds__(256) void ns_div(const float* __restrict__ us, const float* __restrict__ vs,
                                              float* __restrict__ dv) {
    int idx = blockIdx.x * 256 + threadIdx.x;
    int b = idx >> 20, rem = idx & (NSQ - 1);
    int y = rem >> 10, xx = rem & NMSK;
    int xm = (xx + NMSK) & NMSK, xp = (xx + 1) & NMSK;
    int ym = (y + NMSK) & NMSK, yp = (y + 1) & NMSK;
    size_t base = (size_t)b << 20;
    dv[base + (y << 10) + xx] =
        0.5f * (us[base + (y << 10) + xp] - us[base + (y << 10) + xm]) +
        0.5f * (vs[base + (yp << 10) + xx] - vs[base + (ym << 10) + xx]);
}

__global__ __launch_bounds__(256) void ns_scale(float* __restrict__ Gre, float* __restrict__ Gim) {
    int idx = blockIdx.x * 256 + threadIdx.x;
    int rem = idx & (NSQ - 1);
    int m = rem >> 10, n = rem & NMSK;
    float fy = (m <= 512) ? (float)m : (float)(m - 1024);
    float fx = (n <= 512) ? (float)n : (float)(n - 1024);
    float k2 = fmaxf(fy * fy + fx * fx, 1e-10f);
    float sc = (m == 0 && n == 0) ? 0.f : (-1.f / (39.478417604357434f * k2)); // -1/(4*pi^2*k2)
    Gre[idx] *= sc;
    Gim[idx] *= sc;
}

__global__ __launch_bounds__(256) void ns_final(const float* __restrict__ x, const float* __restrict__ us,
                                                const float* __restrict__ vs, const float* __restrict__ phi,
                                                float* __restrict__ out) {
    int idx = blockIdx.x * 256 + threadIdx.x;
    int b = idx >> 20, rem = idx & (NSQ - 1);
    int y = rem >> 10, xx = rem & NMSK;
    int xm = (xx + NMSK) & NMSK, xp = (xx + 1) & NMSK;
    int ym = (y + NMSK) & NMSK, yp = (y + 1) & NMSK;
    size_t base = (size_t)b << 20;
    int c = (y << 10) + xx;
    float px = 0.5f * (phi[base + (y << 10) + xp] - phi[base + (y << 10) + xm]);
    float py = 0.5f * (phi[base + (yp << 10) + xx] - phi[base + (ym << 10) + xx]);
    const float* p = x + (((size_t)b * 3 + 2) << 20);
    size_t ob = ((size_t)b * 3) << 20;
    out[ob + c]            = us[base + c] - px;
    out[ob + NSQ + c]      = vs[base + c] - py;
    out[ob + 2 * NSQ + c]  = p[c] + phi[base + c];
}

// ---------------------------------------------------------------------------
// Batched GEMM via V_WMMA_F32_16X16X4_F32, double-buffered LDS with
// async global->LDS copies (ASYNCcnt) when available.
//   D[z] = alpha * ( A1[z]*B1[z]  +  s2 * A2[z]*B2[z] )    (1024^3, f32)
// Block = 128x128 tile, 8 waves, each wave owns 32x64 (2x4 WMMA tiles).
// A tiles row-major [M][20] (byte-copyable, fragment = one b64 LDS read);
// B tiles [K][132]. Sign s2 folded into A2 fragments (f32 WMMA has no A-NEG).
// ---------------------------------------------------------------------------

__device__ __forceinline__ void ns_copy16(const float* __restrict__ g, float* l) {
#if NS_ASYNC
    // signature (from clang diagnostic): (v4i addrspace(1)*, v4i addrspace(3)*, imm offset, imm cpol)
    __builtin_amdgcn_global_load_async_to_lds_b128(
        (__attribute__((address_space(1))) v4i*)g,
        (__attribute__((address_space(3))) v4i*)l, 0, 0);
#else
    *(float4*)l = *(const float4*)g;
#endif
}

__device__ __forceinline__ void ns_stage(const float* __restrict__ a, const float* __restrict__ b,
                                         float (*Atile)[20], float (*Btile)[132],
                                         int m0, int n0, int k0, int tid) {
#pragma unroll
    for (int s = 0; s < 2; ++s) {              // A: 128x16 f32 = 512 x 16B chunks
        int c = tid + 256 * s;
        int m = c >> 2, ch = (c & 3) * 4;
        ns_copy16(&a[(size_t)(m0 + m) * NN + k0 + ch], &Atile[m][ch]);
    }
#pragma unroll
    for (int s = 0; s < 2; ++s) {              // B: 16x128 f32 = 512 x 16B chunks
        int c = tid + 256 * s;
        int k = c >> 5, ch = (c & 31) * 4;
        ns_copy16(&b[(size_t)(k0 + k) * NN + n0 + ch], &Btile[k][ch]);
    }
}

__device__ __forceinline__ void ns_wmma_tile(const float (*Atile)[20], const float (*Btile)[132],
                                             float sgn, int wm, int wn, int lhi, int llo,
                                             v8f acc[2][4]) {
#pragma unroll
    for (int kk = 0; kk < 16; kk += 4) {
        const int kr = kk + 2 * lhi;
        v2f af[2], bf[4];
#pragma unroll
        for (int i = 0; i < 2; ++i) {
            int m = wm * 32 + i * 16 + llo;
            af[i][0] = Atile[m][kr] * sgn;     // contiguous pair -> ds_load_b64
            af[i][1] = Atile[m][kr + 1] * sgn;
        }
#pragma unroll
        for (int j = 0; j < 4; ++j) {
            int n = wn * 64 + j * 16 + llo;
            bf[j][0] = Btile[kr][n];
            bf[j][1] = Btile[kr + 1][n];
        }
#pragma unroll
        for (int i = 0; i < 2; ++i)
#pragma unroll
            for (int j = 0; j < 4; ++j)
                acc[i][j] = __builtin_amdgcn_wmma_f32_16x16x4_f32(
                    false, af[i], false, bf[j], (short)0, acc[i][j], false, false);
    }
}

__global__ __launch_bounds__(256) void ns_dft_gemm2(
    const float* __restrict__ A1, long long sA1,
    const float* __restrict__ B1, long long sB1,
    const float* __restrict__ A2, long long sA2,
    const float* __restrict__ B2, long long sB2, float s2,
    float* __restrict__ D, float alpha, int has2)
{
    __shared__ alignas(16) float As[2][2][128][20];   // [buf][product][M][Kpad]
    __shared__ alignas(16) float Bs[2][2][16][132];   // [buf][product][K][Npad]

    const int tid  = threadIdx.x;
    const int lane = tid & 31;
    const int lhi  = lane >> 4;
    const int llo  = lane & 15;
    const int wave = tid >> 5;
    const int wm   = wave & 3;      // 4 waves along M
    const int wn   = wave >> 2;     // 2 waves along N
    const int z    = blockIdx.z;
    const int m0   = blockIdx.x * 128;
    const int n0   = blockIdx.y * 128;

    const float* a1 = A1 + (size_t)z * (size_t)sA1;
    const float* b1 = B1 + (size_t)z * (size_t)sB1;
    const float* a2 = has2 ? (A2 + (size_t)z * (size_t)sA2) : a1;
    const float* b2 = has2 ? (B2 + (size_t)z * (size_t)sB2) : b1;

    v8f acc[2][4];
    v8f zv = {0.f, 0.f, 0.f, 0.f, 0.f, 0.f, 0.f, 0.f};
#pragma unroll
    for (int i = 0; i < 2; ++i)
#pragma unroll
        for (int j = 0; j < 4; ++j) acc[i][j] = zv;

    // prologue: stage first K-tile into buffer 0
    ns_stage(a1, b1, As[0][0], Bs[0][0], m0, n0, 0, tid);
    if (has2) ns_stage(a2, b2, As[0][1], Bs[0][1], m0, n0, 0, tid);

    int buf = 0;
    for (int k0 = 0; k0 < NN; k0 += 16) {
        const int nxt = k0 + 16;
        if (nxt < NN) {
            // issue next tile into the other buffer while this tile computes
            ns_stage(a1, b1, As[buf ^ 1][0], Bs[buf ^ 1][0], m0, n0, nxt, tid);
            if (has2) ns_stage(a2, b2, As[buf ^ 1][1], Bs[buf ^ 1][1], m0, n0, nxt, tid);
#if !NS_ASYNC
            __builtin_prefetch(&a1[(size_t)(m0 + (tid >> 1)) * NN + nxt + 16], 0, 0);
            __builtin_prefetch(&b1[(size_t)(nxt + 16 + (tid >> 7)) * NN + n0 + (tid & 127)], 0, 0);
#endif
        }
#if NS_ASYNC
        // ASYNCcnt completes in order: after issuing the next tile's copies,
        // waiting for <= (one tile's issue count) proves the current tile landed.
        if (nxt < NN) {
            if (has2) __builtin_amdgcn_s_wait_asynccnt(8);
            else      __builtin_amdgcn_s_wait_asynccnt(4);
        } else {
            __builtin_amdgcn_s_wait_asynccnt(0);
        }
#endif
        __syncthreads();
        ns_wmma_tile(As[buf][0], Bs[buf][0], 1.f, wm, wn, lhi, llo, acc);
        if (has2) ns_wmma_tile(As[buf][1], Bs[buf][1], s2, wm, wn, lhi, llo, acc);
        __syncthreads();
        buf ^= 1;
    }

    // epilogue: C/D layout -> row = r + 8*lhi within tile, col = llo
    float* Dp = D + (size_t)z * NSQ;
#pragma unroll
    for (int i = 0; i < 2; ++i)
#pragma unroll
        for (int j = 0; j < 4; ++j)
#pragma unroll
            for (int r = 0; r < 8; ++r) {
                int row = m0 + wm * 32 + i * 16 + r + 8 * lhi;
                int col = n0 + wn * 64 + j * 16 + llo;
                Dp[(size_t)row * NN + col] = alpha * acc[i][j][r];
            }
}

// ---------------------------------------------------------------------------
// Orchestration
// ---------------------------------------------------------------------------

extern "C" void kernel_launch(void* const* d_in, const int* in_sizes, int n_in,
                              void* d_out, int out_size, void* d_ws, size_t ws_size,
                              hipStream_t stream) {
    const float* x     = (const float*)d_in[0];
    const float* beta  = (const float*)d_in[1];
    const float* fu_in = (const float*)d_in[2];
    const float* fv_in = (const float*)d_in[3];
    float* out = (float*)d_out;

    char* ws = (char*)d_ws;
    const size_t FSZ = (size_t)TOT * sizeof(float);   // 32 MB field buffer
    const size_t MSZ = (size_t)NSQ * sizeof(float);   // 4 MB DFT matrix
    float* W[8];
    for (int i = 0; i < 8; ++i) W[i] = (float*)(ws + (size_t)i * FSZ);
    float* Cm    = (float*)(ws + 8 * FSZ);
    float* Sm    = (float*)(ws + 8 * FSZ + MSZ);
    float* norms = (float*)(ws + 8 * FSZ + 2 * MSZ);  // 16 floats: [0..7]=|fu|^2, [8..15]=|fv|^2

    const int SBLK = TOT / 256;   // 32768 blocks for field-wide kernels
    const long long SB = NSQ;     // batched stride (elements)

    (void)hipMemsetAsync(norms, 0, 16 * sizeof(float), stream);

    // DFT matrices C,S (F = C - i*S, symmetric)
    ns_gen_dft<<<NSQ / 256, 256, 0, stream>>>(Cm, Sm);

    // 3 smoothing passes: fu: in->W0->W1->W0 ; fv: in->W2->W3->W2
    ns_smooth<<<SBLK, 256, 0, stream>>>(fu_in, W[0], fv_in, W[2]);
    ns_smooth<<<SBLK, 256, 0, stream>>>(W[0], W[1], W[2], W[3]);
    ns_smooth<<<SBLK, 256, 0, stream>>>(W[1], W[0], W[3], W[2]);

    // per-image L2 norms of smoothed forcings
    ns_sumsq<<<dim3(256, BB), 256, 0, stream>>>(W[0], W[2], norms);

    // explicit step -> u*,v* (W4,W5); divergence -> W1
    ns_ustar<<<SBLK, 256, 0, stream>>>(x, beta, W[0], W[2], norms, W[4], W[5]);
    ns_div<<<SBLK, 256, 0, stream>>>(W[4], W[5], W[1]);

    // ---- spectral Poisson solve as batched GEMMs (F = C - iS, symmetric) ----
    dim3 gg(NN / 128, NN / 128, BB);
    const float invN2 = 1.0f / (float)NSQ;
    // forward, left:  P = C*div (W0), Q = S*div (W2)          [T = P - iQ]
    ns_dft_gemm2<<<gg, 256, 0, stream>>>(Cm, 0, W[1], SB, nullptr, 0, nullptr, 0, 0.f, W[0], 1.f, 0);
    ns_dft_gemm2<<<gg, 256, 0, stream>>>(Sm, 0, W[1], SB, nullptr, 0, nullptr, 0, 0.f, W[2], 1.f, 0);
    // forward, right: G_re = P*C - Q*S (W6), G_im = -(P*S + Q*C) (W7, alpha=-1)
    ns_dft_gemm2<<<gg, 256, 0, stream>>>(W[0], SB, Cm, 0, W[2], SB, Sm, 0, -1.f, W[6],  1.f, 1);
    ns_dft_gemm2<<<gg, 256, 0, stream>>>(W[0], SB, Sm, 0, W[2], SB, Cm, 0,  1.f, W[7], -1.f, 1);
    // spectral scale -1/(4pi^2 k^2), zero mean mode
    ns_scale<<<SBLK, 256, 0, stream>>>(W[6], W[7]);
    // inverse, left:  U_re = C*G_re - S*G_im (W0), U_im = S*G_re + C*G_im (W2)
    ns_dft_gemm2<<<gg, 256, 0, stream>>>(Cm, 0, W[6], SB, Sm, 0, W[7], SB, -1.f, W[0], 1.f, 1);
    ns_dft_gemm2<<<gg, 256, 0, stream>>>(Sm, 0, W[6], SB, Cm, 0, W[7], SB,  1.f, W[2], 1.f, 1);
    // inverse, right: phi = (U_re*C - U_im*S)/N^2 (W1)
    ns_dft_gemm2<<<gg, 256, 0, stream>>>(W[0], SB, Cm, 0, W[2], SB, Sm, 0, -1.f, W[1], invN2, 1);

    // projection + output
    ns_final<<<SBLK, 256, 0, stream>>>(x, W[4], W[5], W[1], out);
}